// RandomHingeForest_20057497272597
// MI455X (gfx1250) — compile-verified
//
#include <hip/hip_runtime.h>
#include <cstdint>

typedef __attribute__((ext_vector_type(4))) float f4;

constexpr int BATCH    = 8192;
constexpr int IN_CH    = 1024;
constexpr int TREES    = 1024;
constexpr int DEPTH    = 10;
constexpr int N_NODES  = (1 << DEPTH) - 1;   // 1023 internal nodes per tree
constexpr int N_LEAVES = 1 << DEPTH;         // 1024
constexpr int EXTRA    = 16;
constexpr int TPB      = 256;                // 8 wave32 waves per block
constexpr int TPT      = TREES / TPB;        // 4 trees per thread

__global__ __launch_bounds__(TPB) void hinge_forest_kernel(
    const float* __restrict__ x,     // [BATCH, IN_CH]
    const float* __restrict__ thr,   // [TREES, N_NODES]
    const int*   __restrict__ ord,   // [TREES, N_NODES]
    const float* __restrict__ w,     // [TREES, N_LEAVES, EXTRA]
    float*       __restrict__ out)   // [BATCH, TREES, EXTRA]
{
    __shared__ float xs[IN_CH];      // 4 KB: this sample's feature row

    const int b   = blockIdx.x;
    const int tid = threadIdx.x;

    // --- Stage x[b, :] into LDS via the gfx1250 async DMA path ---------------
    // Each thread issues one GLOBAL_LOAD_ASYNC_TO_LDS_B128 (tracked by
    // ASYNCcnt, no VGPR data movement), then waits its own ASYNCcnt and
    // barriers so all 8 waves see the full row.
    {
        const float* g = x + (size_t)b * IN_CH + (size_t)tid * 4;
        // addrspace(3) offsets live in the low 32 bits of the flat address
        unsigned lds_off = (unsigned)(uintptr_t)(&xs[tid * 4]);
        asm volatile("global_load_async_to_lds_b128 %0, %1, off"
                     :: "v"(lds_off), "v"(g)
                     : "memory");
    }
    asm volatile("s_wait_asynccnt 0" ::: "memory");
    __syncthreads();

    // --- Traverse 4 trees concurrently (ILP across the dependent chain) -----
    int   node[TPT];
    int   tree[TPT];
    float mmin[TPT];
#pragma unroll
    for (int k = 0; k < TPT; ++k) {
        tree[k] = tid + k * TPB;         // lane-contiguous tree ids
        node[k] = 0;
        mmin[k] = __builtin_inff();
    }

#pragma unroll
    for (int d = 0; d < DEPTH; ++d) {
#pragma unroll
        for (int k = 0; k < TPT; ++k) {
            const int   idx = tree[k] * N_NODES + node[k];
            const float t   = thr[idx];          // L2-resident scattered load
            const int   o   = ord[idx];          // L2-resident scattered load
            const float f   = xs[o];             // ds_load_b32 gather from LDS
            const float mg  = f - t;
            mmin[k] = fminf(mmin[k], __builtin_fabsf(mg));
            node[k] = 2 * node[k] + 1 + (mg > 0.0f ? 1 : 0);
        }
    }

    // --- Gather leaf weight rows, scale, stream out (non-temporal) ----------
#pragma unroll
    for (int k = 0; k < TPT; ++k) {
        const int leaf = node[k] - N_NODES;
        const f4* wr = (const f4*)(w + ((size_t)tree[k] * N_LEAVES + leaf) * EXTRA);
        f4*       op = (f4*)(out + ((size_t)b * TREES + tree[k]) * EXTRA);
        const float m = mmin[k];
#pragma unroll
        for (int j = 0; j < 4; ++j) {
            f4 v = wr[j];                        // global_load_b128 (temporal, L2-hot)
            v = v * m;
            __builtin_nontemporal_store(v, op + j);  // global_store_b128 NT: don't
                                                     // evict weights/thr from L2
        }
    }
}

extern "C" void kernel_launch(void* const* d_in, const int* in_sizes, int n_in,
                              void* d_out, int out_size, void* d_ws, size_t ws_size,
                              hipStream_t stream) {
    const float* x   = (const float*)d_in[0];   // [8192, 1024] f32
    const float* thr = (const float*)d_in[1];   // [1024, 1023] f32
    const int*   ord = (const int*)d_in[2];     // [1024, 1023] i32
    const float* w   = (const float*)d_in[3];   // [1024, 1024, 16] f32
    float*       out = (float*)d_out;           // [8192, 1024, 16] f32

    hinge_forest_kernel<<<BATCH, TPB, 0, stream>>>(x, thr, ord, w, out);
}